// BasisVQ_16604343566408
// MI455X (gfx1250) — compile-verified
//
#include <hip/hip_runtime.h>
#include <hip/hip_bf16.h>
#include <math.h>

// Problem sizes (fixed by the reference)
#define M_TOTAL 8192      // B*K = 128*64
#define D_DIM   256       // d_model
#define BD_DIM  2700      // basis_dim
#define NC_DIM  1024      // num_codes
#define N_TILES_ZE 169    // ceil(2700/16)
#define M_TILES 512       // 8192/16
#define KC      540       // K-chunk for dist GEMM: 2700 = 5*540, 540 % 4 == 0

typedef __attribute__((ext_vector_type(2))) float v2f;
typedef __attribute__((ext_vector_type(8))) float v8f;
typedef __attribute__((ext_vector_type(4))) unsigned int u32x4;
typedef __attribute__((ext_vector_type(4))) int i32x4;
typedef __attribute__((ext_vector_type(8))) int i32x8;

__device__ __forceinline__ v8f wmma_f32_16x16x4(v2f a, v2f b, v8f c) {
  // 8 args: (neg_a, A, neg_b, B, c_mod, C, reuse_a, reuse_b)
  return __builtin_amdgcn_wmma_f32_16x16x4_f32(
      false, a, false, b, (short)0, c, false, false);
}

// ---------------------------------------------------------------------------
// TDM 2D tile load: global (tile_x x tile_y, row stride stride_x elements of
// f32) -> LDS at byte offset lds_off. D# layout per CDNA5 ISA ch. 8.
// This toolchain uses the 6-arg builtin: (g0, g1, g2, g3, g4, cpol).
// ---------------------------------------------------------------------------
__device__ __forceinline__ void tdm_load_2d_f32(
    unsigned int lds_off, const float* gsrc,
    unsigned int tile_x, unsigned int tile_y,
    unsigned int tensor_x, unsigned int tensor_y, unsigned int stride_x) {
  unsigned long long ga = (unsigned long long)(uintptr_t)gsrc;
  u32x4 g0;
  g0[0] = 1u;                                       // count=1 (valid user D#)
  g0[1] = lds_off;                                  // lds_addr (bytes)
  g0[2] = (unsigned int)(ga & 0xffffffffu);         // global_addr[31:0]
  g0[3] = (unsigned int)((ga >> 32) & 0x01ffffffu)  // global_addr[56:32]
          | (2u << 30);                             // type=2 ("image")
  i32x8 g1;
  g1[0] = (int)(2u << 16);                          // data_size=2 (4 bytes)
  g1[1] = (int)((tensor_x & 0xffffu) << 16);        // tensor_dim0[15:0] @63:48
  g1[2] = (int)((tensor_x >> 16) |                  // tensor_dim0[31:16] @79:64
                ((tensor_y & 0xffffu) << 16));      // tensor_dim1[15:0] @95:80
  g1[3] = (int)((tensor_y >> 16) |                  // tensor_dim1[31:16]@111:96
                ((tile_x & 0xffffu) << 16));        // tile_dim0 @127:112
  g1[4] = (int)(tile_y & 0xffffu);                  // tile_dim1 @143:128
  g1[5] = (int)stride_x;                            // tensor_dim0_stride[31:0]
  g1[6] = 0;                                        // stride0 hi / stride1 lo
  g1[7] = 0;                                        // stride1 hi
  i32x4 z4 = {0, 0, 0, 0};                          // groups 2/3 unused (2D)
  i32x8 z8 = {0, 0, 0, 0, 0, 0, 0, 0};
  __builtin_amdgcn_tensor_load_to_lds(g0, g1, z4, z4, z8, 0);
}

// ---------------------------------------------------------------------------
// Kernel 1: z_e[8192,2700] = slots[8192,256] @ W[2700,256]^T + bias[2700]
// One wave per 16x16 output tile, f32 WMMA, K-loop of 64 x (K=4).
// ---------------------------------------------------------------------------
__global__ __launch_bounds__(256) void k_gemm_ze(
    const float* __restrict__ A, const float* __restrict__ W,
    const float* __restrict__ bias, float* __restrict__ z_e) {
  int wave = threadIdx.x >> 5;
  int lane = threadIdx.x & 31;
  int lo = lane & 15;
  int hi = lane >> 4;
  int tile = blockIdx.x * 8 + wave;          // 86528 tiles total
  int mt = tile / N_TILES_ZE;
  int nt = tile - mt * N_TILES_ZE;

  int arow = mt * 16 + lo;
  int bcol = nt * 16 + lo;
  bool bvalid = (bcol < BD_DIM);
  const float* aptr = A + (long)arow * D_DIM + 2 * hi;
  const float* bptr = W + (long)(bvalid ? bcol : 0) * D_DIM + 2 * hi;

  v8f acc = {};
#pragma unroll 4
  for (int k0 = 0; k0 < D_DIM; k0 += 4) {
    v2f a = *(const v2f*)(aptr + k0);
    v2f b = *(const v2f*)(bptr + k0);
    if (!bvalid) { b[0] = 0.f; b[1] = 0.f; }
    acc = wmma_f32_16x16x4(a, b, acc);
  }

  int nn = nt * 16 + lo;
  if (nn < BD_DIM) {
    float bb = bias[nn];
#pragma unroll
    for (int v = 0; v < 8; ++v) {
      int mm = mt * 16 + v + 8 * hi;
      z_e[(long)mm * BD_DIM + nn] = acc[v] + bb;
    }
  }
}

// ---------------------------------------------------------------------------
// Kernel 2: per-row squared norms (used for z and for embed)
// ---------------------------------------------------------------------------
__global__ __launch_bounds__(256) void k_rownorm(
    const float* __restrict__ src, int cols, float* __restrict__ out) {
  int row = blockIdx.x;
  const float* p = src + (long)row * cols;
  float s = 0.f;
  for (int c = threadIdx.x; c < cols; c += 256) { float v = p[c]; s += v * v; }
  __shared__ float red[8];
  for (int off = 16; off; off >>= 1) s += __shfl_xor(s, off, 32);
  if ((threadIdx.x & 31) == 0) red[threadIdx.x >> 5] = s;
  __syncthreads();
  if (threadIdx.x < 32) {
    float t = (threadIdx.x < 8) ? red[threadIdx.x] : 0.f;
    for (int off = 4; off; off >>= 1) t += __shfl_xor(t, off, 32);
    if (threadIdx.x == 0) out[row] = t;
  }
}

// ---------------------------------------------------------------------------
// Kernel 3 (fused): for a 16-row chunk, compute dist[16,1024] via f32 WMMA
// into LDS, then per-row argmin + stable softmax + avg_probs atomics.
// The shared 16-row z_e panel is staged through LDS by the Tensor Data
// Mover (tensor_load_to_lds) in K-chunks of 540, so the 8 waves read it
// once from HBM instead of 8x redundantly.
// ---------------------------------------------------------------------------
__global__ __launch_bounds__(256) void k_dist_argmin_softmax(
    const float* __restrict__ z_e, const float* __restrict__ E,
    const float* __restrict__ znorm, const float* __restrict__ enorm,
    float* __restrict__ avg_probs, int* __restrict__ idx_out,
    float* __restrict__ ind_out) {
  __shared__ float ldist[16 * NC_DIM];   // 64 KB
  __shared__ float lA[16 * KC];          // 34.6 KB staged z_e panel
  int wave = threadIdx.x >> 5;
  int lane = threadIdx.x & 31;
  int lo = lane & 15;
  int hi = lane >> 4;
  int blk = blockIdx.x;                  // 0..511

  const float* bbase = E + 2 * hi;
  int ncol0 = wave * 128 + lo;           // (wave*8 + t)*16 + lo, t folded below

#pragma unroll
  for (int t = 0; t < 8; ++t)
    __builtin_prefetch(bbase + (long)(ncol0 + t * 16) * BD_DIM, 0, 0);

  const float* lAp = lA + lo * KC + 2 * hi;  // A fragment base in LDS
  unsigned int lA_off = (unsigned int)(uintptr_t)(void*)lA;

  v8f acc[8] = {};
  for (int kc = 0; kc < BD_DIM; kc += KC) {
    __syncthreads();                     // previous panel fully consumed
    if (wave == 0) {
      // DMA 16 x KC tile of z_e (row stride BD_DIM) into LDS
      tdm_load_2d_f32(lA_off, z_e + (long)blk * 16 * BD_DIM + kc,
                      /*tile_x=*/KC, /*tile_y=*/16,
                      /*tensor_x=*/BD_DIM, /*tensor_y=*/M_TOTAL,
                      /*stride_x=*/BD_DIM);
      __builtin_amdgcn_s_wait_tensorcnt(0);
    }
    __syncthreads();                     // panel visible to all waves

    for (int k0 = 0; k0 < KC; k0 += 4) {
      v2f a = *(const v2f*)(lAp + k0);   // ds_load_b64 from staged panel
#pragma unroll
      for (int t = 0; t < 8; ++t) {
        v2f b = *(const v2f*)(bbase + (long)(ncol0 + t * 16) * BD_DIM + kc + k0);
        acc[t] = wmma_f32_16x16x4(a, b, acc[t]);
      }
    }
  }

  float zn[8];
#pragma unroll
  for (int v = 0; v < 8; ++v) zn[v] = znorm[blk * 16 + v + 8 * hi];
#pragma unroll
  for (int t = 0; t < 8; ++t) {
    int nn = ncol0 + t * 16;
    float en = enorm[nn];
#pragma unroll
    for (int v = 0; v < 8; ++v) {
      int mm = v + 8 * hi;
      ldist[mm * NC_DIM + nn] = zn[v] - 2.f * acc[t][v] + en;
    }
  }
  __syncthreads();

  // Each wave reduces 2 rows of the 16-row chunk.
  for (int rr = 0; rr < 2; ++rr) {
    int r = wave * 2 + rr;
    const float* drow = &ldist[r * NC_DIM];

    float mind = 3.4e38f; int mini = 0;
    for (int c = lane; c < NC_DIM; c += 32) {
      float d = drow[c];
      if (d < mind || (d == mind && c < mini)) { mind = d; mini = c; }
    }
    for (int off = 16; off; off >>= 1) {
      float od = __shfl_xor(mind, off, 32);
      int   oi = __shfl_xor(mini, off, 32);
      if (od < mind || (od == mind && oi < mini)) { mind = od; mini = oi; }
    }

    float s = 0.f;
    for (int c = lane; c < NC_DIM; c += 32) s += __expf(mind - drow[c]);
    for (int off = 16; off; off >>= 1) s += __shfl_xor(s, off, 32);
    float inv = 1.f / (s * (float)M_TOTAL);

    for (int c = lane; c < NC_DIM; c += 32)
      atomicAdd(&avg_probs[c], __expf(mind - drow[c]) * inv);

    if (lane == 0) {
      int row = blk * 16 + r;
      idx_out[row] = mini;
      ind_out[row] = (float)mini;
    }
  }
}

// ---------------------------------------------------------------------------
// Kernel 4: q_st = embed[idx] gather; accumulate vq_loss = sum (z_e - e)^2
// ---------------------------------------------------------------------------
__global__ __launch_bounds__(256) void k_gather_vq(
    const float* __restrict__ E, const float* __restrict__ z_e,
    const int* __restrict__ idx, float* __restrict__ q_st,
    float* __restrict__ vq_accum) {
  int row = blockIdx.x;
  int code = idx[row];
  const float* ep = E + (long)code * BD_DIM;
  const float* zp = z_e + (long)row * BD_DIM;
  float* qp = q_st + (long)row * BD_DIM;
  float s = 0.f;
  for (int c = threadIdx.x; c < BD_DIM; c += 256) {
    float e = ep[c];
    qp[c] = e;
    float dz = zp[c] - e;
    s += dz * dz;
  }
  __shared__ float red[8];
  for (int off = 16; off; off >>= 1) s += __shfl_xor(s, off, 32);
  if ((threadIdx.x & 31) == 0) red[threadIdx.x >> 5] = s;
  __syncthreads();
  if (threadIdx.x < 32) {
    float t = (threadIdx.x < 8) ? red[threadIdx.x] : 0.f;
    for (int off = 4; off; off >>= 1) t += __shfl_xor(t, off, 32);
    if (threadIdx.x == 0) atomicAdd(vq_accum, t);
  }
}

// ---------------------------------------------------------------------------
// Kernel 5: entropy over avg_probs + scalar outputs
// ---------------------------------------------------------------------------
__global__ __launch_bounds__(1024) void k_finalize(
    const float* __restrict__ avg_probs, const float* __restrict__ vq_accum,
    float* __restrict__ out_scalars) {
  __shared__ float red[32];
  int tid = threadIdx.x;                   // 1024 threads
  float v = avg_probs[tid];
  float h = -v * logf(v + 1e-8f);
  for (int off = 16; off; off >>= 1) h += __shfl_xor(h, off, 32);
  if ((tid & 31) == 0) red[tid >> 5] = h;
  __syncthreads();
  if (tid < 32) {
    float t = red[tid];
    for (int off = 16; off; off >>= 1) t += __shfl_xor(t, off, 32);
    if (tid == 0) {
      out_scalars[0] = vq_accum[0] * (1.f / (float)((long)M_TOTAL * BD_DIM));
      out_scalars[1] = t;
    }
  }
}

__global__ __launch_bounds__(256) void k_zero(float* __restrict__ p, int n) {
  int i = blockIdx.x * 256 + threadIdx.x;
  if (i < n) p[i] = 0.f;
}

// ---------------------------------------------------------------------------
// Workspace layout (floats):
//   [0)   z_e (22,118,400) | znorm (8192) | enorm (1024) | avg_probs (1024)
//         | vq_accum (1) | idx as int (8192)
// d_out layout: q_st (22,118,400) | indices-as-float (8192) | vq_loss | entropy
// ---------------------------------------------------------------------------
extern "C" void kernel_launch(void* const* d_in, const int* in_sizes, int n_in,
                              void* d_out, int out_size, void* d_ws, size_t ws_size,
                              hipStream_t stream) {
  const float* slots = (const float*)d_in[0];   // [128,64,256]
  const float* W     = (const float*)d_in[1];   // [2700,256]
  const float* bias  = (const float*)d_in[2];   // [2700]
  const float* embed = (const float*)d_in[3];   // [1024,2700]

  float* ws = (float*)d_ws;
  const long ZE = 0;
  const long ZN = ZE + (long)M_TOTAL * BD_DIM;  // 22,118,400
  const long EN = ZN + M_TOTAL;
  const long AP = EN + NC_DIM;
  const long VQ = AP + NC_DIM;
  const long IX = VQ + 1;

  float* z_e       = ws + ZE;
  float* znorm     = ws + ZN;
  float* enorm     = ws + EN;
  float* avg_probs = ws + AP;
  float* vq_accum  = ws + VQ;
  int*   idxbuf    = (int*)(ws + IX);

  float* q_st    = (float*)d_out;
  float* ind_out = q_st + (long)M_TOTAL * BD_DIM;
  float* scalars = ind_out + M_TOTAL;

  // zero avg_probs + vq_accum (d_ws is poisoned, not re-zeroed by harness)
  k_zero<<<5, 256, 0, stream>>>(avg_probs, NC_DIM + 1);

  // z_e GEMM: 512*169 tiles, 8 waves/block
  k_gemm_ze<<<(M_TILES * N_TILES_ZE) / 8, 256, 0, stream>>>(slots, W, bias, z_e);

  // norms
  k_rownorm<<<M_TOTAL, 256, 0, stream>>>(z_e, BD_DIM, znorm);
  k_rownorm<<<NC_DIM, 256, 0, stream>>>(embed, BD_DIM, enorm);

  // fused dist + argmin + softmax (512 blocks of 16 rows)
  k_dist_argmin_softmax<<<M_TILES, 256, 0, stream>>>(
      z_e, embed, znorm, enorm, avg_probs, idxbuf, ind_out);

  // gather + vq loss
  k_gather_vq<<<M_TOTAL, 256, 0, stream>>>(embed, z_e, idxbuf, q_st, vq_accum);

  // entropy + scalars
  k_finalize<<<1, 1024, 0, stream>>>(avg_probs, vq_accum, scalars);
}